// NSETransformer_45509473469109
// MI455X (gfx1250) — compile-verified
//
#include <hip/hip_runtime.h>

namespace {
constexpr int Dv = 96, Hv = 128, Wv = 160;
constexpr int Nv = Dv * Hv * Wv;                 // 1,966,080
constexpr float MXI = 1.0f / 159.0f;             // 1 / (max(SIZE)-1)
}

__global__ __launch_bounds__(160)
void warp3d_kernel(const float* __restrict__ src,    // (D,H,W,2)
                   const float* __restrict__ flow,   // (N,7): t0..2, qx,qy,qz,qw
                   float* __restrict__ out)          // [2N warped][3N new_loc][3N grid]
{
    // Block = one W-row (160 voxels, 5 waves). Flow chunk = 1120 floats = 4480 B.
    __shared__ __align__(16) float sflow[Wv * 7];

    const int tid  = threadIdx.x;                 // x coordinate (vector)
    const int yq   = blockIdx.x;                  // y coordinate (scalar)
    const int zq   = blockIdx.y;                  // z coordinate (scalar)
    const int base = (zq * Hv + yq) * Wv;         // scalar voxel base of this row
    const int n    = base + tid;

    // ---- block-cooperative async stage: 280 x b128 = 4480 B of flow into LDS ----
    {
        const float* gbase = flow + (size_t)base * 7;   // 16B aligned (base % 160 == 0)
        unsigned     la0 = (unsigned)(size_t)(&sflow[tid * 4]);
        const float* ga0 = gbase + tid * 4;
        asm volatile("global_load_async_to_lds_b128 %0, %1, off"
                     :: "v"(la0), "v"(ga0) : "memory");
        if (tid < 280 - 160) {                          // second round: slots 160..279
            unsigned     la1 = (unsigned)(size_t)(&sflow[(tid + 160) * 4]);
            const float* ga1 = gbase + (tid + 160) * 4;
            asm volatile("global_load_async_to_lds_b128 %0, %1, off"
                         :: "v"(la1), "v"(ga1) : "memory");
        }
        asm volatile("s_wait_asynccnt 0x0" ::: "memory");  // own DMA landed
        __syncthreads();                                   // every wave's DMA landed
    }

    // ---- grid point (reference: 2*(i/(s-1)-0.5)*((s-1)/mx)); p0,p1 are scalar ----
    const float p0 = 2.0f * ((float)zq  * (1.0f / (Dv - 1)) - 0.5f) * ((float)(Dv - 1) * MXI);
    const float p1 = 2.0f * ((float)yq  * (1.0f / (Hv - 1)) - 0.5f) * ((float)(Hv - 1) * MXI);
    const float p2 = 2.0f * ((float)tid * (1.0f / (Wv - 1)) - 0.5f) * ((float)(Wv - 1) * MXI);

    // ---- flow from LDS (stride-7 dwords: gcd(7,64)=1 -> conflict-free) ----
    const float* f  = sflow + tid * 7;
    const float  t0 = f[0], t1 = f[1], t2 = f[2];
    const float  qx = f[3], qy = f[4], qz = f[5], qw = f[6];

    // uv = cross(qv, p) + qw*p ; new_loc = p + 2*cross(qv, uv) + t
    const float u0 = qy * p2 - qz * p1 + qw * p0;
    const float u1 = qz * p0 - qx * p2 + qw * p1;
    const float u2 = qx * p1 - qy * p0 + qw * p2;

    const float nl0 = p0 + 2.0f * (qy * u2 - qz * u1) + t0;
    const float nl1 = p1 + 2.0f * (qz * u0 - qx * u2) + t1;
    const float nl2 = p2 + 2.0f * (qx * u1 - qy * u0) + t2;

    // ---- sampling coords: new_locs = (new_loc/scale)[..., ::-1] ----
    const float xs = nl2;                            // scale[2] == 1
    const float ys = nl1 * (159.0f / 127.0f);
    const float zs = nl0 * (159.0f / 95.0f);

    const float ix = fminf(fmaxf((xs + 1.0f) * 0.5f * (float)(Wv - 1), 0.0f), (float)(Wv - 1));
    const float iy = fminf(fmaxf((ys + 1.0f) * 0.5f * (float)(Hv - 1), 0.0f), (float)(Hv - 1));
    const float iz = fminf(fmaxf((zs + 1.0f) * 0.5f * (float)(Dv - 1), 0.0f), (float)(Dv - 1));

    const int x0 = (int)floorf(ix); const int x1 = min(x0 + 1, Wv - 1);
    const int y0 = (int)floorf(iy); const int y1 = min(y0 + 1, Hv - 1);
    const int z0 = (int)floorf(iz); const int z1 = min(z0 + 1, Dv - 1);

    const float wx = ix - (float)x0, wy = iy - (float)y0, wz = iz - (float)z0;
    const float wx0 = 1.0f - wx, wy0 = 1.0f - wy, wz0 = 1.0f - wz;

    // ---- 8 corner gathers (b64, both channels); src = 15.7 MB -> resident in L2 ----
    const float2* v = reinterpret_cast<const float2*>(src);
    const int b00 = (z0 * Hv + y0) * Wv, b01 = (z0 * Hv + y1) * Wv;
    const int b10 = (z1 * Hv + y0) * Wv, b11 = (z1 * Hv + y1) * Wv;

    const float2 c000 = v[b00 + x0], c001 = v[b00 + x1];
    const float2 c010 = v[b01 + x0], c011 = v[b01 + x1];
    const float2 c100 = v[b10 + x0], c101 = v[b10 + x1];
    const float2 c110 = v[b11 + x0], c111 = v[b11 + x1];

    const float w000 = wz0 * wy0 * wx0, w001 = wz0 * wy0 * wx;
    const float w010 = wz0 * wy  * wx0, w011 = wz0 * wy  * wx;
    const float w100 = wz  * wy0 * wx0, w101 = wz  * wy0 * wx;
    const float w110 = wz  * wy  * wx0, w111 = wz  * wy  * wx;

    const float a0 = w000 * c000.x + w001 * c001.x + w010 * c010.x + w011 * c011.x
                   + w100 * c100.x + w101 * c101.x + w110 * c110.x + w111 * c111.x;
    const float a1 = w000 * c000.y + w001 * c001.y + w010 * c010.y + w011 * c011.y
                   + w100 * c100.y + w101 * c101.y + w110 * c110.y + w111 * c111.y;

    // ---- 8 coalesced N-strided output streams (saddr + tid addressing) ----
    out[0 * Nv + n] = a0;     // warped ch0
    out[1 * Nv + n] = a1;     // warped ch1
    out[2 * Nv + n] = nl0;    // new_loc (channel-first)
    out[3 * Nv + n] = nl1;
    out[4 * Nv + n] = nl2;
    out[5 * Nv + n] = p0;     // grid (channel-first)
    out[6 * Nv + n] = p1;
    out[7 * Nv + n] = p2;
}

extern "C" void kernel_launch(void* const* d_in, const int* in_sizes, int n_in,
                              void* d_out, int out_size, void* d_ws, size_t ws_size,
                              hipStream_t stream) {
    const float* src  = (const float*)d_in[0];   // (1,D,H,W,2) float32
    const float* flow = (const float*)d_in[1];   // (N,7) float32
    float*       out  = (float*)d_out;           // 8N float32

    dim3 grid(Hv, Dv);                           // (128, 96) rows
    dim3 block(Wv);                              // 160 threads = 5 waves
    warp3d_kernel<<<grid, block, 0, stream>>>(src, flow, out);
}